// GraphPooling_41394894799378
// MI455X (gfx1250) — compile-verified
//
#include <hip/hip_runtime.h>

typedef unsigned int u32;
typedef unsigned long long u64;

typedef unsigned int u32x4 __attribute__((ext_vector_type(4)));
typedef int          i32x8 __attribute__((ext_vector_type(8)));
typedef int          i32x4 __attribute__((ext_vector_type(4)));

#define VCAP  (1u<<20)   // voxel hash capacity (M <= N=400k -> load factor <= 0.39)
#define ECAP  (1u<<23)   // edge hash capacity (K <= E=3.2M -> load factor <= 0.39)
#define ESORT (1u<<22)   // edge sort buffer, pow2 >= E
#define TROWS 128        // nodes per block staged through LDS by the TDM
#define LROW  65         // LDS row stride in DWORDs (64 data + 1 pad from TDM pad_*)
#define EMPTY64 0xFFFFFFFFFFFFFFFFull

static __device__ __forceinline__ u64 mix64(u64 x){
  x ^= x>>33; x *= 0xff51afd7ed558ccdull;
  x ^= x>>33; x *= 0xc4ceb9fe1a85ec53ull;
  x ^= x>>33; return x;
}
// Order-preserving u32 encoding of f32: unsigned compare == float compare.
// Lets segment-max use one native GLOBAL_ATOMIC_MAX_U32 per element (no CAS loop).
// Identity element is 0 (below every real float's encoding).
static __device__ __forceinline__ u32 encf(float f){
  u32 u = __float_as_uint(f);
  return (u & 0x80000000u) ? ~u : (u | 0x80000000u);
}
static __device__ __forceinline__ float decf(u32 u){
  return __uint_as_float((u & 0x80000000u) ? (u & 0x7FFFFFFFu) : ~u);
}

// ---------------- init workspace (re-run every call: deterministic) ----------
__global__ void k_init(u32* qmax, u32* Mctr, u32* Kctr, u64* vkey, u32* vid,
                       u64* skey, float* psum, u32* pcnt, u64* ekey, u64* elist, int N){
  u32 i = blockIdx.x*blockDim.x + threadIdx.x;
  if (i < 3) qmax[i] = 0u;
  if (i == 0){ *Mctr = 0u; *Kctr = 0u; }
  if (i < VCAP){ vkey[i] = EMPTY64; vid[i] = 0u; skey[i] = EMPTY64; }
  if (i < (u32)N){ pcnt[i]=0u; psum[3u*i]=0.f; psum[3u*i+1u]=0.f; psum[3u*i+2u]=0.f; }
  if (i < ECAP)  ekey[i]  = EMPTY64;
  if (i < ESORT) elist[i] = EMPTY64;
}

// ---------------- pass 1: per-dim max of floor(pos/4) (strides) --------------
__global__ void k_qmax(const float* pos, int N, u32* qmax){
  int n = blockIdx.x*blockDim.x + threadIdx.x;
  if (n >= N) return;
  #pragma unroll
  for (int c = 0; c < 3; ++c){
    int q = (int)floorf(pos[3*n+c] * 0.25f);          // pos in [0,128): q >= 0
    atomicMax(&qmax[c], (u32)q);
  }
}

// ---------------- pass 2: voxel key -> hash insert, arrival ids --------------
__global__ void k_insert(const float* pos, const int* batch, int N, const u32* qmax,
                         u64* vkey, u32* vid, u32* inva, u64* skey, u32* Mctr){
  int n = blockIdx.x*blockDim.x + threadIdx.x;
  if (n >= N) return;
  u64 s0 = qmax[0]+1u, s1 = qmax[1]+1u, s2 = qmax[2]+1u;
  u64 k = (u64)(u32)batch[n];
  k = k*s0 + (u64)(u32)(int)floorf(pos[3*n+0]*0.25f);
  k = k*s1 + (u64)(u32)(int)floorf(pos[3*n+1]*0.25f);
  k = k*s2 + (u64)(u32)(int)floorf(pos[3*n+2]*0.25f);
  u32 h = (u32)mix64(k) & (VCAP-1u);
  u32 slot;
  for (;;){                                     // claim loop: no waiting on others
    u64 old = atomicCAS(&vkey[h], EMPTY64, k);  // GLOBAL_ATOMIC_CMPSWAP_B64
    if (old == EMPTY64){
      u32 id = atomicAdd(Mctr, 1u);
      skey[id] = k;                             // unique keys in arrival order
      __hip_atomic_store(&vid[h], id+1u, __ATOMIC_RELEASE, __HIP_MEMORY_SCOPE_AGENT);
      slot = h; break;
    }
    if (old == k){ slot = h; break; }
    h = (h+1u) & (VCAP-1u);
  }
  u32 v;                                        // producer already past its CAS -> resident
  do { v = __hip_atomic_load(&vid[slot], __ATOMIC_ACQUIRE, __HIP_MEMORY_SCOPE_AGENT); }
  while (v == 0u);
  inva[n] = v - 1u;
}

// ---------------- bitonic sort step (ascending; EMPTY64 sorts last) ----------
__global__ void k_bitonic(u64* d, u32 n, u32 k, u32 j){
  u32 i = blockIdx.x*blockDim.x + threadIdx.x;
  u32 ixj = i ^ j;
  if (i < n && ixj > i){
    bool up = ((i & k) == 0u);
    u64 a = d[i], b = d[ixj];
    if ((a > b) == up){ d[i] = b; d[ixj] = a; }
  }
}

// ---------------- arrival id -> sorted rank (matches jnp.unique order) -------
__global__ void k_rank(const u64* skey_sorted, const u64* vkey, const u32* vid,
                       u32* rank, const u32* Mctr){
  u32 j = blockIdx.x*blockDim.x + threadIdx.x;
  if (j >= *Mctr) return;
  u64 k = skey_sorted[j];
  u32 h = (u32)mix64(k) & (VCAP-1u);
  while (vkey[h] != k) h = (h+1u) & (VCAP-1u);
  rank[vid[h]-1u] = j;
}

__global__ void k_invf(const u32* inva, const u32* rank, u32* invf, int N){
  int n = blockIdx.x*blockDim.x + threadIdx.x;
  if (n < N) invf[n] = rank[inva[n]];
}

// zero the encoded pooled_x region of d_out (identity for encoded max)
__global__ void k_initout(u32* outb, const u32* Mctr){
  u32 i = blockIdx.x*blockDim.x + threadIdx.x;
  if (i < (*Mctr)*64u) outb[i] = 0u;
}

// ---------------- segment mean of pos: native f32 add atomics ----------------
__global__ void k_pos(const float* pos, const u32* invf, int N, float* psum, u32* pcnt){
  int n = blockIdx.x*blockDim.x + threadIdx.x;
  if (n >= N) return;
  u32 f = invf[n];
  atomicAdd(&psum[3u*(size_t)f+0u], pos[3*n+0]);   // GLOBAL_ATOMIC_ADD_F32
  atomicAdd(&psum[3u*(size_t)f+1u], pos[3*n+1]);
  atomicAdd(&psum[3u*(size_t)f+2u], pos[3*n+2]);
  atomicAdd(&pcnt[f], 1u);
}

// ---------------- segment max of x, staged through LDS by the TDM ------------
// TDM loads a TROWS x 64 f32 tile; pad_interval=64 DWORDs / pad_amount=1 DWORD
// gives a 65-DWORD LDS row stride -> lane t reads bank (t+f)%64: conflict-free.
__global__ __launch_bounds__(TROWS) void k_xmax(const float* x, const u32* invf,
                                                int N, float* outp){
  __shared__ float tile[TROWS*LROW];           // only LDS object -> offset 0
  int start = blockIdx.x * TROWS;
  int rows  = min(TROWS, N - start);
  // Opaque zero (scalar asm): compiler cannot fold it, so the dummy pad-column
  // store below and the +zr load indices keep tile[] reads un-foldable while
  // still being direct addrspace(3) accesses -> real ds_store/ds_load.
  u32 zr; asm volatile("s_mov_b32 %0, 0" : "=s"(zr));
  // Store into the pad column (col 64). The TDM skips pad DWORDs (address
  // increment only, no write), so this never races with the DMA payload.
  tile[threadIdx.x*LROW + 64u + zr] = 0.0f;    // ds_store_b32
#if __has_builtin(__builtin_amdgcn_tensor_load_to_lds) && __has_builtin(__builtin_amdgcn_s_wait_tensorcnt)
  if (threadIdx.x < 32u){                      // one wave issues the DMA (EXEC ignored)
    u64 ga = (u64)(const void*)(x + (size_t)start*64u);
    u32x4 g0;
    g0[0] = 1u;                                // count=1 valid descriptor
    g0[1] = 0u;                                // lds_addr = 0 (tile[] base)
    g0[2] = (u32)ga;                           // global_addr[31:0]
    g0[3] = (u32)((ga>>32) & 0x1FFFFFFu) | (2u<<30); // addr[56:32], type=2 (image)
    i32x8 g1;
    g1[0] = (int)((2u<<16) | (1u<<20) | (5u<<22));   // data_size=4B, pad_en, interval=64dw, amount=1dw
    g1[1] = (int)(64u<<16);                    // tensor_dim0 = 64 elements
    g1[2] = (int)(((u32)rows & 0xFFFFu)<<16);  // tensor_dim1 = rows
    g1[3] = (int)(64u<<16);                    // tile_dim0 = 64
    g1[4] = (int)((u32)rows & 0xFFFFu);        // tile_dim1 = rows, tile_dim2 = 0
    g1[5] = 64;                                // tensor_dim0_stride = 64 elements
    g1[6] = 0; g1[7] = 0;
    i32x4 z4 = {0,0,0,0};                      // groups 2/3 unused (2D tensor)
    i32x8 z8 = {0,0,0,0,0,0,0,0};              // extra group (6-arg toolchain form)
    __builtin_amdgcn_tensor_load_to_lds(g0, g1, z4, z4, z8, 0);
    __builtin_amdgcn_s_wait_tensorcnt(0);      // s_wait_tensorcnt 0
  }
  __syncthreads();
#else
  for (int i = threadIdx.x; i < rows*64; i += blockDim.x)
    tile[(i>>6)*LROW + (i&63)] = x[(size_t)start*64u + i];
  __syncthreads();
#endif
  u32 t = threadIdx.x;
  if ((int)t < rows){
    u32 fid = invf[start + (int)t];
    u32* dst = (u32*)outp + (size_t)fid*64u;
    #pragma unroll
    for (u32 f = 0; f < 64u; ++f)
      atomicMax(&dst[f], encf(tile[t*LROW + f + zr]));  // ds_load_b32 + GLOBAL_ATOMIC_MAX_U32
  }
}

__global__ void k_decode(float* out, const u32* Mctr){
  u32 i = blockIdx.x*blockDim.x + threadIdx.x;
  if (i < (*Mctr)*64u) out[i] = decf(((const u32*)out)[i]);
}

__global__ void k_posmean(const float* psum, const u32* pcnt, const u32* Mctr, float* out){
  u32 j = blockIdx.x*blockDim.x + threadIdx.x;
  u32 M = *Mctr;
  if (j >= M) return;
  float inv = 1.0f / (float)pcnt[j];
  size_t base = (size_t)M*64u + 3u*(size_t)j;
  out[base+0] = psum[3u*(size_t)j+0u]*inv;
  out[base+1] = psum[3u*(size_t)j+1u]*inv;
  out[base+2] = psum[3u*(size_t)j+2u]*inv;
}

// ---------------- edge remap + dedup (u64 hash) ------------------------------
__global__ void k_edges(const int* ei, int E, const u32* invf, const u32* Mctr,
                        u64* ekey, u64* elist, u32* Kctr){
  int e = blockIdx.x*blockDim.x + threadIdx.x;
  if (e >= E) return;
  __builtin_prefetch(ei + 2*(e + 4096), 0, 0);        // global_prefetch_b8
  u32 fa = invf[ei[2*e]], fb = invf[ei[2*e+1]];
  if (fa == fb) return;                               // drop self-loops
  u64 M = (u64)*Mctr;
  u64 enc = (u64)fa*M + (u64)fb;                      // < M^2 << 2^63: never EMPTY64
  u32 h = (u32)mix64(enc) & (ECAP-1u);
  for (;;){
    u64 old = atomicCAS(&ekey[h], EMPTY64, enc);
    if (old == EMPTY64){ elist[atomicAdd(Kctr,1u)] = enc; return; }
    if (old == enc) return;
    h = (h+1u) & (ECAP-1u);
  }
}

__global__ void k_edges_out(const u64* elist, const u32* Mctr, const u32* Kctr, float* out){
  u32 i = blockIdx.x*blockDim.x + threadIdx.x;
  u64 M = (u64)*Mctr; u32 K = *Kctr;
  size_t base = (size_t)M*67u;                        // after pooled_x + pos_mean
  if (i < K){
    u64 e = elist[i];
    out[base + 2u*(size_t)i]      = (float)(e / M);
    out[base + 2u*(size_t)i + 1u] = (float)(e % M);
  } else if (i < K + (u32)M){
    float v = (float)(i - K);                         // appended self-loops
    out[base + 2u*(size_t)i]      = v;
    out[base + 2u*(size_t)i + 1u] = v;
  }
}

__global__ void k_batch(const u64* skey_sorted, const u32* qmax, const u32* Mctr,
                        const u32* Kctr, float* out){
  u32 j = blockIdx.x*blockDim.x + threadIdx.x;
  u32 M = *Mctr;
  if (j >= M) return;
  u64 P = (u64)(qmax[0]+1u)*(u64)(qmax[1]+1u)*(u64)(qmax[2]+1u);
  size_t base = (size_t)M*67u + 2u*((size_t)*Kctr + (size_t)M);
  out[base + j] = (float)(skey_sorted[j] / P);
}

// ---------------- host orchestration ----------------------------------------
extern "C" void kernel_launch(void* const* d_in, const int* in_sizes, int n_in,
                              void* d_out, int out_size, void* d_ws, size_t ws_size,
                              hipStream_t stream){
  const float* x     = (const float*)d_in[0];
  const float* pos   = (const float*)d_in[1];
  const int*   ei    = (const int*)d_in[2];   // integer inputs arrive as int32
  const int*   batch = (const int*)d_in[3];
  const int N = in_sizes[0] / 64;
  const int E = in_sizes[2] / 2;
  float* out = (float*)d_out;

  char* w = (char*)d_ws;
  auto take = [&](size_t bytes)->char*{ char* p = w; w += (bytes + 255u) & ~(size_t)255u; return p; };
  u32*   qmax = (u32*)take(16);
  u32*   Mctr = (u32*)take(16);
  u32*   Kctr = (u32*)take(16);
  u64*   vkey = (u64*)take((size_t)VCAP*8u);
  u32*   vid  = (u32*)take((size_t)VCAP*4u);
  u64*   skey = (u64*)take((size_t)VCAP*8u);
  u32*   inva = (u32*)take((size_t)N*4u);
  u32*   invf = (u32*)take((size_t)N*4u);
  u32*   rank = (u32*)take((size_t)N*4u);
  float* psum = (float*)take((size_t)N*12u);
  u32*   pcnt = (u32*)take((size_t)N*4u);
  u64*   ekey = (u64*)take((size_t)ECAP*8u);
  u64*   elist= (u64*)take((size_t)ESORT*8u);   // total ~140 MB of d_ws

  const u32 B = 256u;
  const u32 gN  = ((u32)N + B-1u)/B;
  const u32 gE  = ((u32)E + B-1u)/B;
  const u32 gV  = VCAP/B;
  const u32 gX  = ((u32)N*64u + B-1u)/B;
  const u32 gI  = ECAP/B;                        // init covers largest region
  const u32 gEo = (ESORT + (u32)N + B-1u)/B;

  k_init   <<<gI, B, 0, stream>>>(qmax, Mctr, Kctr, vkey, vid, skey, psum, pcnt, ekey, elist, N);
  k_qmax   <<<gN, B, 0, stream>>>(pos, N, qmax);
  k_insert <<<gN, B, 0, stream>>>(pos, batch, N, qmax, vkey, vid, inva, skey, Mctr);
  for (u32 k = 2u; k <= VCAP; k <<= 1)           // sort unique voxel keys ascending
    for (u32 j = k>>1; j > 0u; j >>= 1)
      k_bitonic<<<gV, B, 0, stream>>>(skey, VCAP, k, j);
  k_rank   <<<gV, B, 0, stream>>>(skey, vkey, vid, rank, Mctr);
  k_invf   <<<gN, B, 0, stream>>>(inva, rank, invf, N);
  k_initout<<<gX, B, 0, stream>>>((u32*)d_out, Mctr);
  k_pos    <<<gN, B, 0, stream>>>(pos, invf, N, psum, pcnt);
  k_xmax   <<<((u32)N + TROWS-1u)/TROWS, TROWS, 0, stream>>>(x, invf, N, out);
  k_decode <<<gX, B, 0, stream>>>(out, Mctr);
  k_posmean<<<gV, B, 0, stream>>>(psum, pcnt, Mctr, out);
  k_edges  <<<gE, B, 0, stream>>>(ei, E, invf, Mctr, ekey, elist, Kctr);
  for (u32 k = 2u; k <= ESORT; k <<= 1)          // sort deduped edges ascending
    for (u32 j = k>>1; j > 0u; j >>= 1)
      k_bitonic<<<ESORT/B, B, 0, stream>>>(elist, ESORT, k, j);
  k_edges_out<<<gEo, B, 0, stream>>>(elist, Mctr, Kctr, out);
  k_batch  <<<gV, B, 0, stream>>>(skey, qmax, Mctr, Kctr, out);
}